// Residual_74629351735838
// MI455X (gfx1250) — compile-verified
//
#include <hip/hip_runtime.h>
#include <cstdint>
#include <cstddef>

// ---------------------------------------------------------------------------
// Bundle-adjustment reprojection residual for MI455X (gfx1250).
//
// Memory-bound kernel (~1 FLOP/byte): all optimization effort goes into
// the data-movement path — streaming arrays use wide NT loads/stores, the
// observation stream is staged via the CDNA5 async global->LDS pipe
// (ASYNCcnt), and pose/point gathers hit the 192MB L2 (tables are 56KB/6MB).
// ---------------------------------------------------------------------------

typedef float f32x2_t __attribute__((ext_vector_type(2)));
typedef float f32x4_t __attribute__((ext_vector_type(4)));
typedef int   i32x2_t __attribute__((ext_vector_type(2)));

#define THREADS 256

struct KMat { float m00,m01,m02,m10,m11,m12,m20,m21,m22; };

struct Gather { float tx,ty,tz,qx,qy,qz,qw,px,py,pz; };

__device__ __forceinline__ Gather load_gather(const float* __restrict__ poses,
                                              const float* __restrict__ points,
                                              int c, int pt) {
  Gather g;
  const float* P = poses + (size_t)c * 7;   // 28B record, L1/L2 resident
  g.tx = P[0]; g.ty = P[1]; g.tz = P[2];
  g.qx = P[3]; g.qy = P[4]; g.qz = P[5]; g.qw = P[6];
  const float* X = points + (size_t)pt * 3; // 12B record, L2 resident
  g.px = X[0]; g.py = X[1]; g.pz = X[2];
  return g;
}

__device__ __forceinline__ f32x2_t residual(const Gather& g, const KMat& K,
                                            float ox, float oy) {
  // uv = qv x p
  float uvx = g.qy*g.pz - g.qz*g.py;
  float uvy = g.qz*g.px - g.qx*g.pz;
  float uvz = g.qx*g.py - g.qy*g.px;
  // s = uv + qw*p
  float sx = uvx + g.qw*g.px;
  float sy = uvy + g.qw*g.py;
  float sz = uvz + g.qw*g.pz;
  // w = qv x s
  float wx = g.qy*sz - g.qz*sy;
  float wy = g.qz*sx - g.qx*sz;
  float wz = g.qx*sy - g.qy*sx;
  // p_cam = p + 2w + t
  float pcx = g.px + 2.0f*wx + g.tx;
  float pcy = g.py + 2.0f*wy + g.ty;
  float pcz = g.pz + 2.0f*wz + g.tz;
  // img = K * p_cam (general 3x3, FMA-contracted)
  float ix = K.m00*pcx + K.m01*pcy + K.m02*pcz;
  float iy = K.m10*pcx + K.m11*pcy + K.m12*pcz;
  float iz = K.m20*pcx + K.m21*pcy + K.m22*pcz;
  float inv = 1.0f / iz;
  f32x2_t r;
  r.x = ix*inv - ox;
  r.y = iy*inv - oy;
  return r;
}

__global__ __launch_bounds__(THREADS) void ba_residual_kernel(
    const float* __restrict__ observes,
    const float* __restrict__ Km,
    const float* __restrict__ poses,
    const float* __restrict__ points,
    const int*  __restrict__ cidx,
    const int*  __restrict__ pidx,
    float* __restrict__ out,
    int n_obs)
{
  __shared__ float lds_obs[THREADS * 4];   // 4KB: 2 obs (4 floats) per thread

  const int pair = blockIdx.x * THREADS + threadIdx.x;
  const int i0   = pair * 2;

  // K is uniform & read-only: clang promotes these to SMEM scalar loads.
  KMat K;
  K.m00 = Km[0]; K.m01 = Km[1]; K.m02 = Km[2];
  K.m10 = Km[3]; K.m11 = Km[4]; K.m12 = Km[5];
  K.m20 = Km[6]; K.m21 = Km[7]; K.m22 = Km[8];

  if (i0 + 1 < n_obs) {
    // --- CDNA5 async global->LDS: stage this thread's observation pair
    // (16B, perfectly coalesced across the wave) while the random pose /
    // point gathers below are in flight. Tracked by ASYNCcnt.
    unsigned ldsoff   = (unsigned)(uintptr_t)&lds_obs[threadIdx.x * 4];
    const float* gobs = observes + (size_t)i0 * 2;
    asm volatile("global_load_async_to_lds_b128 %0, %1, off"
                 :: "v"(ldsoff), "v"(gobs) : "memory");

    // Streaming index loads: 8B each, non-temporal (read once; keep L2 for
    // the pose/point tables).
    i32x2_t c2 = __builtin_nontemporal_load((const i32x2_t*)(cidx + i0));
    i32x2_t p2 = __builtin_nontemporal_load((const i32x2_t*)(pidx + i0));

    // Issue both gathers before waiting on the async copy so their latency
    // overlaps with the LDS staging.
    Gather g0 = load_gather(poses, points, c2.x, p2.x);
    Gather g1 = load_gather(poses, points, c2.y, p2.y);

    asm volatile("s_wait_asynccnt 0x0" ::: "memory");
    float ox0 = lds_obs[threadIdx.x*4 + 0];
    float oy0 = lds_obs[threadIdx.x*4 + 1];
    float ox1 = lds_obs[threadIdx.x*4 + 2];
    float oy1 = lds_obs[threadIdx.x*4 + 3];

    f32x2_t r0 = residual(g0, K, ox0, oy0);
    f32x2_t r1 = residual(g1, K, ox1, oy1);

    // 16B non-temporal store of both residual pairs (write-once stream).
    f32x4_t res; res.x = r0.x; res.y = r0.y; res.z = r1.x; res.w = r1.y;
    __builtin_nontemporal_store(res, (f32x4_t*)(out + (size_t)i0 * 2));
  } else if (i0 < n_obs) {
    // Tail (at most one odd observation).
    float ox = observes[(size_t)i0*2 + 0];
    float oy = observes[(size_t)i0*2 + 1];
    Gather g = load_gather(poses, points, cidx[i0], pidx[i0]);
    f32x2_t r = residual(g, K, ox, oy);
    out[(size_t)i0*2 + 0] = r.x;
    out[(size_t)i0*2 + 1] = r.y;
  }
}

extern "C" void kernel_launch(void* const* d_in, const int* in_sizes, int n_in,
                              void* d_out, int out_size, void* d_ws, size_t ws_size,
                              hipStream_t stream) {
  // setup_inputs() order: observes, K, poses, points, cidx, pidx
  const float* observes = (const float*)d_in[0];
  const float* K        = (const float*)d_in[1];
  const float* poses    = (const float*)d_in[2];
  const float* points   = (const float*)d_in[3];
  const int*   cidx     = (const int*)d_in[4];
  const int*   pidx     = (const int*)d_in[5];
  float*       out      = (float*)d_out;

  const int n_obs = in_sizes[0] / 2;           // observes is (N_OBS, 2)
  const int pairs = (n_obs + 1) / 2;
  const int blocks = (pairs + THREADS - 1) / THREADS;

  ba_residual_kernel<<<dim3(blocks), dim3(THREADS), 0, stream>>>(
      observes, K, poses, points, cidx, pidx, out, n_obs);
}